// HypAgg_29240137351644
// MI455X (gfx1250) — compile-verified
//
#include <hip/hip_runtime.h>

// ---------------------------------------------------------------------------
// HypAgg (Poincare-ball graph attention aggregation) for MI455X / gfx1250.
//
// Roofline: x (25.6MB) and x_t (25.6MB) both fit in the 192MB L2, so the
// ~1.2GB of random per-edge gathers are L2-resident; HBM traffic is ~100MB
// (trivial at 23.3 TB/s). The kernel is L2-bandwidth / f32-atomic bound.
// FLOPs are nearly free, so we compute the 16 per-edge dot products of a
// wave's edge batch with 16x V_WMMA_F32_16X16X4_F32 (matrix pipe), extract
// the diagonal with ds_bpermute, and keep the VALU free for the
// artanh/exp/tanh transcendental chain.
// ---------------------------------------------------------------------------

typedef float v2f __attribute__((ext_vector_type(2)));
typedef float v8f __attribute__((ext_vector_type(8)));

#define DIMS 64
#define MAXNORM 0.996f      /* (1 - PROJ_EPS)/sqrt(c), c = 1 */
#define MIN_NORM 1e-15f

__device__ __forceinline__ float artanh_clip(float v) {
  float u = fminf(v, 1.f - 1e-7f);
  u = fmaxf(u, -1.f + 1e-7f);
  return 0.5f * logf((1.f + u) / (1.f - u));
}

// edge weight from the three scalars (c == 1):
//  ma = mobius_add(-p1, p2): |num|^2 = A^2 x2 + B^2 y2 - 2AB (p1.p2),
//  A = 1 - 2 p1.p2 + y2, B = 1 - x2, denom = 1 - 2 p1.p2 + x2 y2
__device__ __forceinline__ float edge_weight(float dot, float x2, float y2,
                                             float beta, float con) {
  float A  = 1.f - 2.f * dot + y2;
  float B  = 1.f - x2;
  float n2 = A * A * x2 + B * B * y2 - 2.f * A * B * dot;
  float de = fmaxf(1.f - 2.f * dot + x2 * y2, MIN_NORM);
  float ma = sqrtf(fmaxf(n2, 0.f)) / de;
  float dist = 2.f * artanh_clip(ma);
  return expf(fmaf(-beta, dist * dist, con));
}

// --------------------------- node preparation ------------------------------
// 16 lanes per node (float4 each): sqn[i] = |x_i|^2, xt[i] = logmap0(x_i).
__global__ void hypagg_prep(const float* __restrict__ x,
                            float* __restrict__ xt,
                            float* __restrict__ sqn, int N) {
  int t = blockIdx.x * blockDim.x + threadIdx.x;
  int node = t >> 4;
  int sub  = t & 15;
  if (node >= N) return;
  const float4* px = reinterpret_cast<const float4*>(x + (long long)node * DIMS);
  float4 p = px[sub];
  float s = p.x * p.x + p.y * p.y + p.z * p.z + p.w * p.w;
  // reduce across the aligned 16-lane group (wave32; masks stay in-group)
  s += __shfl_xor(s, 1, 32);
  s += __shfl_xor(s, 2, 32);
  s += __shfl_xor(s, 4, 32);
  s += __shfl_xor(s, 8, 32);
  if (sub == 0) sqn[node] = s;
  float pn    = fmaxf(sqrtf(s), MIN_NORM);
  float scale = artanh_clip(pn) / pn;   // logmap0, c = 1
  float4 o;
  o.x = p.x * scale; o.y = p.y * scale; o.z = p.z * scale; o.w = p.w * scale;
  reinterpret_cast<float4*>(xt + (long long)node * DIMS)[sub] = o;
}

// ------------------------------ edge kernel --------------------------------
// One wave handles 16 edges per iteration.  Dot products via
// V_WMMA_F32_16X16X4_F32: A = 16 edges' p1 rows (16xK), B = 16 edges' p2
// columns (Kx16); the diagonal of the 16x16 result is the per-edge dot.
// ISA 7.12.2 32-bit A 16x4 layout: lanes 0-15 hold M=0..15; VGPR0 = K{0,2}
// (lo/hi half-wave), VGPR1 = K{1,3}.  B assumed symmetric (per-column).
__global__ void __launch_bounds__(256)
hypagg_edges(const float* __restrict__ x,
             const float* __restrict__ xt,
             const float* __restrict__ sqn,
             const long long* __restrict__ ei,   // [2,E] int64
             const float* __restrict__ betap,
             const float* __restrict__ conp,
             float* __restrict__ support,        // d_out accumulator [N,64]
             float* __restrict__ rowsum,         // [N]
             int E, int ngroups) {
  const int lane = threadIdx.x & 31;
  const int eL   = lane & 15;          // which edge of the 16-batch
  const int hi   = lane >> 4;          // half-wave -> K sub-pair {0,1}/{2,3}
  const int wib  = threadIdx.x >> 5;
  const int wpb  = blockDim.x >> 5;
  const long long gwave  = (long long)blockIdx.x * wpb + wib;
  const long long nwaves = (long long)gridDim.x * wpb;
  const float beta = betap[0];
  const float con  = conp[0];

  for (long long g = gwave; g < (long long)ngroups; g += nwaves) {
    const long long base = g * 16;
    int r = (int)ei[base + eL];
    int c = (int)ei[(long long)E + base + eL];

    // speculative prefetch of the next batch's indices (global_prefetch_b8)
    long long nb = base + nwaves * 16;
    if (nb < (long long)E) {
      __builtin_prefetch(ei + nb + eL, 0, 1);
      __builtin_prefetch(ei + (long long)E + nb + eL, 0, 1);
    }

    const float* p1 = x + (long long)r * DIMS;
    const float* p2 = x + (long long)c * DIMS;
    const float2* a2 = reinterpret_cast<const float2*>(p1 + 2 * hi);
    const float2* b2 = reinterpret_cast<const float2*>(p2 + 2 * hi);

    v8f acc = {};
#pragma unroll
    for (int k = 0; k < 16; ++k) {       // 16 K=4 chunks cover D=64
      float2 ta = a2[2 * k];             // p1[4k+2hi .. +1]
      float2 tb = b2[2 * k];             // p2[4k+2hi .. +1]
      v2f av; av.x = ta.x; av.y = ta.y;
      v2f bv; bv.x = tb.x; bv.y = tb.y;
      acc = __builtin_amdgcn_wmma_f32_16x16x4_f32(
          false, av, false, bv, (short)0, acc, false, false);
    }

    // Diagonal extraction: element (M=e,N=e) lives in VGPR (e&7),
    // lane e (e<8) or lane e+16 (e>=8).  Gather with ds_bpermute.
    int srcLane = (eL < 8) ? eL : (eL + 16);
    int jsel    = eL & 7;
    float dot = 0.f;
#pragma unroll
    for (int j = 0; j < 8; ++j) {
      float tj = __shfl(acc[j], srcLane, 32);
      dot = (j == jsel) ? tj : dot;
    }

    // per-edge attention weight (duplicated in both half-waves)
    float w = edge_weight(dot, sqn[r], sqn[c], beta, con);

    // scatter: all 32 lanes cover 64 dims (2 each) per edge
    const float2* xtb = reinterpret_cast<const float2*>(xt);
#pragma unroll 1
    for (int e = 0; e < 16; ++e) {
      float we = __shfl(w, e, 32);
      int   rr = __shfl(r, e, 32);
      int   cc = __shfl(c, e, 32);
      float2 v = xtb[(long long)cc * (DIMS / 2) + lane];
      float* dst = support + (long long)rr * DIMS + 2 * lane;
      atomicAdd(dst,     we * v.x);
      atomicAdd(dst + 1, we * v.y);
      if (lane == 0) atomicAdd(rowsum + rr, we);
    }
  }
}

// scalar tail for E % 16 (not hit for E = 1.6M, kept for generality)
__global__ void hypagg_tail(const float* __restrict__ x,
                            const float* __restrict__ xt,
                            const float* __restrict__ sqn,
                            const long long* __restrict__ ei,
                            const float* __restrict__ betap,
                            const float* __restrict__ conp,
                            float* __restrict__ support,
                            float* __restrict__ rowsum,
                            int E, int start) {
  int e = start + blockIdx.x * blockDim.x + threadIdx.x;
  if (e >= E) return;
  int r = (int)ei[e];
  int c = (int)ei[(long long)E + e];
  const float* p1 = x + (long long)r * DIMS;
  const float* p2 = x + (long long)c * DIMS;
  float dot = 0.f;
  for (int i = 0; i < DIMS; ++i) dot = fmaf(p1[i], p2[i], dot);
  float w = edge_weight(dot, sqn[r], sqn[c], betap[0], conp[0]);
  const float* s = xt + (long long)c * DIMS;
  float* d = support + (long long)r * DIMS;
  for (int i = 0; i < DIMS; ++i) atomicAdd(d + i, w * s[i]);
  atomicAdd(rowsum + r, w);
}

// ------------------------------- finalize ----------------------------------
// out = proj(expmap0(support / (rowsum + 1e-10))), 16 lanes per node.
__global__ void hypagg_finalize(float* __restrict__ out,
                                const float* __restrict__ rowsum, int N) {
  int t = blockIdx.x * blockDim.x + threadIdx.x;
  int node = t >> 4;
  int sub  = t & 15;
  if (node >= N) return;
  float4* po = reinterpret_cast<float4*>(out + (long long)node * DIMS);
  float4 s = po[sub];
  float inv = 1.f / (rowsum[node] + 1e-10f);
  s.x *= inv; s.y *= inv; s.z *= inv; s.w *= inv;
  float n2 = s.x * s.x + s.y * s.y + s.z * s.z + s.w * s.w;
  n2 += __shfl_xor(n2, 1, 32);
  n2 += __shfl_xor(n2, 2, 32);
  n2 += __shfl_xor(n2, 4, 32);
  n2 += __shfl_xor(n2, 8, 32);
  float sn = sqrtf(n2);
  float un = fmaxf(sn, MIN_NORM);        // expmap0 norm clamp
  float th = tanhf(un);
  float sc = th / un;
  s.x *= sc; s.y *= sc; s.z *= sc; s.w *= sc;
  float resn = th * (sn / un);           // |expmap0(s)|
  float pn = fmaxf(resn, MIN_NORM);
  if (pn > MAXNORM) {
    float k = MAXNORM / pn;
    s.x *= k; s.y *= k; s.z *= k; s.w *= k;
  }
  po[sub] = s;
}

// ------------------------------- launcher ----------------------------------
extern "C" void kernel_launch(void* const* d_in, const int* in_sizes, int n_in,
                              void* d_out, int out_size, void* d_ws, size_t ws_size,
                              hipStream_t stream) {
  const float*     x    = (const float*)d_in[0];
  const float*     beta = (const float*)d_in[1];
  const float*     con  = (const float*)d_in[2];
  const long long* ei   = (const long long*)d_in[3];   // int64 [2,E]
  const int N = in_sizes[0] / DIMS;
  const int E = in_sizes[3] / 2;

  // workspace: xt [N*64] | sqn [N] | rowsum [N]   (~26.4 MB for N=100k)
  float* xt     = (float*)d_ws;
  float* sqn    = xt + (size_t)N * DIMS;
  float* rowsum = sqn + N;

  hipMemsetAsync(d_out,  0, (size_t)N * DIMS * sizeof(float), stream);
  hipMemsetAsync(rowsum, 0, (size_t)N * sizeof(float), stream);

  int prepThreads = N * 16;
  hypagg_prep<<<(prepThreads + 255) / 256, 256, 0, stream>>>(x, xt, sqn, N);

  int ngroups = E / 16;
  if (ngroups > 0) {
    int blocks = (ngroups + 7) / 8;          // 8 waves per 256-thread block
    if (blocks > 16384) blocks = 16384;      // grid-stride beyond this
    hypagg_edges<<<blocks, 256, 0, stream>>>(x, xt, sqn, ei, beta, con,
                                             (float*)d_out, rowsum, E, ngroups);
  }
  int tail = E - ngroups * 16;
  if (tail > 0) {
    hypagg_tail<<<(tail + 63) / 64, 64, 0, stream>>>(x, xt, sqn, ei, beta, con,
                                                     (float*)d_out, rowsum,
                                                     E, ngroups * 16);
  }
  hypagg_finalize<<<(prepThreads + 255) / 256, 256, 0, stream>>>((float*)d_out,
                                                                 rowsum, N);
}